// GCN_90744069030459
// MI455X (gfx1250) — compile-verified
//
#include <hip/hip_runtime.h>
#include <hip/hip_bf16.h>

typedef __attribute__((ext_vector_type(2))) float    v2f;
typedef __attribute__((ext_vector_type(8))) float    v8f;
typedef __attribute__((ext_vector_type(4))) unsigned tdm_v4u;
typedef __attribute__((ext_vector_type(8))) int      tdm_v8i;
typedef __attribute__((ext_vector_type(4))) int      tdm_v4i;

#define DIN 128

#if defined(__gfx1250__) && __has_builtin(__builtin_amdgcn_tensor_load_to_lds)
#define USE_TDM 1
#else
#define USE_TDM 0
#endif

// ---------------------------------------------------------------- utilities
__global__ void zero_f32_kernel(float* __restrict__ p, long long n) {
    long long i = (long long)blockIdx.x * blockDim.x + threadIdx.x;
    long long stride = (long long)gridDim.x * blockDim.x;
    for (; i < n; i += stride) p[i] = 0.0f;
}

__global__ void count_deg_kernel(const int* __restrict__ dst, float* __restrict__ deg, int E) {
    int e = blockIdx.x * blockDim.x + threadIdx.x;
    if (e < E) atomicAdd(&deg[dst[e]], 1.0f);
}

__global__ void dinv_kernel(const float* __restrict__ deg, float* __restrict__ dinv, int n) {
    int i = blockIdx.x * blockDim.x + threadIdx.x;
    if (i < n) dinv[i] = rsqrtf(deg[i] + 1.0f);   // +1 = self loop; deg >= 1 always
}

// ------------------------------------------------------ WMMA fp32 GEMM
// T[nrows, NOUT] = X[nrows, 128] @ W[128, NOUT]
// One wave computes a 16x16 output tile via V_WMMA_F32_16X16X4_F32 (K-loop of 32).
// The 16x128 X tile is DMA'd into LDS by the Tensor Data Mover (wave 0), with
// hardware zero-fill for out-of-range tail rows (tensor_dim1 = nrows - row0).
template<int NOUT>
__global__ __launch_bounds__(32 * (NOUT / 16))
void gemm_wmma_f32(const float* __restrict__ X, const float* __restrict__ W,
                   float* __restrict__ T, int nrows) {
    __shared__ float xs[16 * DIN];          // 8 KB
    const int row0 = blockIdx.x * 16;
    const int tid  = threadIdx.x;

#if USE_TDM
    if (tid < 32) {
        const unsigned long long ga =
            (unsigned long long)(const void*)X + (unsigned long long)row0 * DIN * 4ull;
        const unsigned lds_off = (unsigned)(unsigned long long)(void*)xs;
        const unsigned remain  = (unsigned)(nrows - row0);   // rows available; >=1

        // D# group 0: count=1 | lds_addr | global_addr[56:0] | type=2
        tdm_v4u g0;
        g0[0] = 1u;                                   // count=1, is_restore=0
        g0[1] = lds_off;                              // LDS byte address
        g0[2] = (unsigned)ga;                         // global_addr[31:0]
        g0[3] = (unsigned)((ga >> 32) & 0x01ffffffull) | (2u << 30);  // [56:32] | type=2

        // D# group 1: data_size=4B | tensor_dim0=128 | tensor_dim1=remain |
        //             tile_dim0=128 | tile_dim1=16 | dim0_stride=128
        tdm_v8i g1;
        g1[0] = (int)(2u << 16);                      // data_size code 2 (4 bytes)
        g1[1] = (int)(128u << 16);                    // tensor_dim0[15:0] = 128
        g1[2] = (int)((remain & 0xffffu) << 16);      // tensor_dim0 hi=0 | tensor_dim1[15:0]
        g1[3] = (int)(((remain >> 16) & 0xffffu) | (128u << 16)); // tensor_dim1 hi | tile_dim0=128
        g1[4] = 16;                                   // tile_dim1=16 | tile_dim2=0
        g1[5] = 128;                                  // tensor_dim0_stride[31:0] = 128
        g1[6] = 0;                                    // stride hi | tensor_dim1_stride lo
        g1[7] = 0;

        tdm_v4i gz4 = {0, 0, 0, 0};                   // groups 2/3: dims unused
        tdm_v8i gz8 = {0, 0, 0, 0, 0, 0, 0, 0};
        __builtin_amdgcn_tensor_load_to_lds(g0, g1, gz4, gz4, gz8, 0);
        __builtin_amdgcn_s_wait_tensorcnt(0);
    }
    __syncthreads();
#else
    {
        const int nthr = 32 * (NOUT / 16);
        for (int i = tid; i < 16 * DIN; i += nthr) {
            int r = i >> 7, c = i & (DIN - 1);
            int gr = row0 + r;
            xs[i] = (gr < nrows) ? X[(long long)gr * DIN + c] : 0.0f;
        }
        __syncthreads();
    }
#endif

    const int wave = tid >> 5;
    const int lane = tid & 31;
    const int n0   = wave * 16;
    const int lm   = lane & 15;   // M for A-frag, N for B/C-frag
    const int lh   = lane >> 4;   // selects K pair / row half

    v8f acc = {};
    for (int k = 0; k < DIN; k += 4) {
        const int ka = k + lh * 2;
        v2f a;                                   // A: 16x4, lane lm = row, VGPR j -> K = ka+j
        a.x = xs[lm * DIN + ka + 0];
        a.y = xs[lm * DIN + ka + 1];
        v2f b;                                   // B: 4x16, lane lm = col, VGPR j -> K = ka+j
        b.x = W[(ka + 0) * NOUT + n0 + lm];
        b.y = W[(ka + 1) * NOUT + n0 + lm];
        acc = __builtin_amdgcn_wmma_f32_16x16x4_f32(
                false, a, false, b, (short)0, acc, false, false);
    }

    // C/D layout: VGPR r -> M = r + 8*lh, N = lm
    for (int r = 0; r < 8; ++r) {
        int gr = row0 + r + lh * 8;
        if (gr < nrows) T[(long long)gr * NOUT + n0 + lm] = acc[r];
    }
}

// ------------------------------------- epilogue init: agg = T*dinv^2 + bias
__global__ void init_agg_kernel(const float* __restrict__ T, const float* __restrict__ bias,
                                const float* __restrict__ dinv, float* __restrict__ agg,
                                int n, int D) {
    long long idx = (long long)blockIdx.x * blockDim.x + threadIdx.x;
    if (idx >= (long long)n * D) return;
    int node = (int)(idx / D);
    int f    = (int)(idx - (long long)node * D);
    float di = dinv[node];
    agg[idx] = T[idx] * di * di + bias[f];
}

// ----------------------------------------------- edge gather/scale/scatter
// one thread = one (edge, 4-feature chunk); a wave of 32 covers one 128-wide edge
__global__ void scatter_edges_kernel(const float* __restrict__ T,
                                     const int* __restrict__ src,
                                     const int* __restrict__ dst,
                                     const float* __restrict__ dinv,
                                     float* __restrict__ agg,
                                     int E, int D) {
    const int chunks = D >> 2;
    long long idx = (long long)blockIdx.x * blockDim.x + threadIdx.x;
    long long total = (long long)E * chunks;
    if (idx >= total) return;
    int e = (int)(idx / chunks);
    int c = (int)(idx - (long long)e * chunks);
    int s = src[e], d = dst[e];
    float nrm = dinv[s] * dinv[d];
    const float4 v = ((const float4*)(T + (long long)s * D))[c];
    float* ap = agg + (long long)d * D + (c << 2);
    atomicAdd(ap + 0, v.x * nrm);
    atomicAdd(ap + 1, v.y * nrm);
    atomicAdd(ap + 2, v.z * nrm);
    atomicAdd(ap + 3, v.w * nrm);
}

__global__ void relu_kernel(float* __restrict__ p, long long n) {
    long long idx = (long long)blockIdx.x * blockDim.x + threadIdx.x;
    if (idx < n) p[idx] = fmaxf(p[idx], 0.0f);
}

// ---------------------------------------------------------------- driver
extern "C" void kernel_launch(void* const* d_in, const int* in_sizes, int n_in,
                              void* d_out, int out_size, void* d_ws, size_t ws_size,
                              hipStream_t stream) {
    const float* x   = (const float*)d_in[0];
    const int*   ei  = (const int*)  d_in[1];
    const float* W1  = (const float*)d_in[2];
    const float* b1  = (const float*)d_in[3];
    const float* W2  = (const float*)d_in[4];
    const float* b2  = (const float*)d_in[5];
    const float* W3  = (const float*)d_in[6];
    const float* b3  = (const float*)d_in[7];

    const int N = in_sizes[0] / DIN;
    const int E = in_sizes[1] / 2;
    const int* src = ei;
    const int* dst = ei + E;

    // d_out layout: y [N*64] | h1 [N*128] | h2 [N*128]
    float* y  = (float*)d_out;
    float* h1 = y  + (long long)N * 64;
    float* h2 = h1 + (long long)N * 128;

    // workspace: deg[N] | dinv[N] | T[N*128]
    float* deg  = (float*)d_ws;
    float* dinv = deg + N;
    float* T    = dinv + N;

    auto zgrid = [](long long n) { long long b = (n + 255) / 256; return (int)(b > 65535 ? 65535 : b); };

    // ---- degrees / normalization
    zero_f32_kernel<<<zgrid(N), 256, 0, stream>>>(deg, N);
    count_deg_kernel<<<(E + 255) / 256, 256, 0, stream>>>(dst, deg, E);
    dinv_kernel<<<(N + 255) / 256, 256, 0, stream>>>(deg, dinv, N);

    const int gblocks = (N + 15) / 16;
    const long long nd128 = (long long)N * 128;
    const long long nd64  = (long long)N * 64;
    const long long eth128 = (long long)E * 32;   // E * (128/4)
    const long long eth64  = (long long)E * 16;   // E * (64/4)

    // ---- layer 1: h1 = relu(agg(x @ W1) + b1)
    gemm_wmma_f32<128><<<gblocks, 256, 0, stream>>>(x, W1, T, N);
    init_agg_kernel<<<(int)((nd128 + 255) / 256), 256, 0, stream>>>(T, b1, dinv, h1, N, 128);
    scatter_edges_kernel<<<(int)((eth128 + 255) / 256), 256, 0, stream>>>(T, src, dst, dinv, h1, E, 128);
    relu_kernel<<<(int)((nd128 + 255) / 256), 256, 0, stream>>>(h1, nd128);

    // ---- layer 2: h2 = relu(agg(h1 @ W2) + b2)
    gemm_wmma_f32<128><<<gblocks, 256, 0, stream>>>(h1, W2, T, N);
    init_agg_kernel<<<(int)((nd128 + 255) / 256), 256, 0, stream>>>(T, b2, dinv, h2, N, 128);
    scatter_edges_kernel<<<(int)((eth128 + 255) / 256), 256, 0, stream>>>(T, src, dst, dinv, h2, E, 128);
    relu_kernel<<<(int)((nd128 + 255) / 256), 256, 0, stream>>>(h2, nd128);

    // ---- layer 3: y = agg(h2 @ W3) + b3   (no relu -> no extra pass)
    gemm_wmma_f32<64><<<gblocks, 128, 0, stream>>>(h2, W3, T, N);
    init_agg_kernel<<<(int)((nd64 + 255) / 256), 256, 0, stream>>>(T, b3, dinv, y, N, 64);
    scatter_edges_kernel<<<(int)((eth64 + 255) / 256), 256, 0, stream>>>(T, src, dst, dinv, y, E, 64);
}